// DGN_75411035783778
// MI455X (gfx1250) — compile-verified
//
#include <hip/hip_runtime.h>

// DGN (3-layer GCN + pooling) for MI455X / gfx1250, wave32.
// GEMM uses V_WMMA_F32_16X16X4_F32 (full fp32, matches reference precision).
// Edge scatter/pool use atomics; h/agg (25.6MB each) are L2-resident (192MB L2).
// Streaming passes vectorized to b128; scatter is one wave per edge (float2/lane).

#define DIM 64
#define NG  64   // number of graphs

typedef float v2f __attribute__((ext_vector_type(2)));
typedef float v8f __attribute__((ext_vector_type(8)));

// ---------------------------------------------------------------------------
// init: deg[i]=1 (self-loop), counts=0, graph-embedding output = 0
__global__ void k_init(float* __restrict__ deg, float* __restrict__ counts,
                       float* __restrict__ gout, int n) {
    int i = blockIdx.x * blockDim.x + threadIdx.x;
    if (i < n) deg[i] = 1.0f;
    if (i < NG) counts[i] = 0.0f;
    if (i < 2 * NG * DIM) gout[i] = 0.0f;
}

// degree accumulation over destination nodes
__global__ void k_deg(const int* __restrict__ dst, float* __restrict__ deg, int e) {
    int i = blockIdx.x * blockDim.x + threadIdx.x;
    if (i < e) atomicAdd(&deg[dst[i]], 1.0f);
}

// dis = rsqrt(deg) in place; also count nodes per graph
__global__ void k_dis_counts(float* __restrict__ deg_dis, const int* __restrict__ batch,
                             float* __restrict__ counts, int n) {
    int i = blockIdx.x * blockDim.x + threadIdx.x;
    if (i < n) {
        float d = deg_dis[i];
        deg_dis[i] = (d > 0.0f) ? rsqrtf(d) : 0.0f;
        atomicAdd(&counts[batch[i]], 1.0f);
    }
}

// ---------------------------------------------------------------------------
// H = X @ W  via V_WMMA_F32_16X16X4_F32.
// Block = 256 threads = 8 waves. Each wave owns one 16-row strip of X and
// computes all 4 column tiles (N = 64), reusing the A-tile registers 4x.
//
// A (16x4 f32, ISA 7.12.2): lane L: M = L%16, holds K = (L/16)*2 + {0,1}
//   -> one float2 load per lane per K-step.
// B (4x16): row (K) striped across lanes in a VGPR: v[j] = W[k + half*2 + j][n0 + L%16]
// C/D (16x16): VGPR r: lanes 0-15 -> M=r, lanes 16-31 -> M=r+8.
__global__ __launch_bounds__(256) void k_gemm(const float* __restrict__ X,
                                              const float* __restrict__ W,
                                              float* __restrict__ H, int n) {
    const int lane   = threadIdx.x & 31;
    const int wave   = threadIdx.x >> 5;
    const int mstrip = blockIdx.x * 8 + wave;          // 16-row strip index
    if (mstrip >= n / 16) return;                      // wave-uniform exit (EXEC all-1 for WMMA)
    const int m0   = mstrip * 16;
    const int half = lane >> 4;                        // 0: lanes 0-15, 1: lanes 16-31
    const int l16  = lane & 15;
    const size_t arow = (size_t)(m0 + l16) * DIM;

    v8f c0 = {}, c1 = {}, c2 = {}, c3 = {};
    #pragma unroll
    for (int k = 0; k < DIM; k += 4) {
        // A tile: K = k + half*2 + {0,1}
        v2f a = *(const v2f*)(X + arow + k + half * 2);
        const float* wrow0 = W + (size_t)(k + half * 2 + 0) * DIM + l16;
        const float* wrow1 = W + (size_t)(k + half * 2 + 1) * DIM + l16;
        v2f b;
        b.x = wrow0[0];  b.y = wrow1[0];
        c0 = __builtin_amdgcn_wmma_f32_16x16x4_f32(false, a, false, b, (short)0, c0, false, false);
        b.x = wrow0[16]; b.y = wrow1[16];
        c1 = __builtin_amdgcn_wmma_f32_16x16x4_f32(false, a, false, b, (short)0, c1, false, false);
        b.x = wrow0[32]; b.y = wrow1[32];
        c2 = __builtin_amdgcn_wmma_f32_16x16x4_f32(false, a, false, b, (short)0, c2, false, false);
        b.x = wrow0[48]; b.y = wrow1[48];
        c3 = __builtin_amdgcn_wmma_f32_16x16x4_f32(false, a, false, b, (short)0, c3, false, false);
    }
    #pragma unroll
    for (int r = 0; r < 8; ++r) {
        const size_t row = (size_t)(m0 + r + half * 8) * DIM + l16;
        H[row +  0] = c0[r];
        H[row + 16] = c1[r];
        H[row + 32] = c2[r];
        H[row + 48] = c3[r];
    }
}

// ---------------------------------------------------------------------------
// agg = h * dis^2   (self-loop contribution doubles as the zero-init).
// float4 per thread -> global_load_b128 / global_store_b128.
__global__ void k_self(const float4* __restrict__ H4, const float* __restrict__ dis,
                       float4* __restrict__ agg4, int n) {
    int t = blockIdx.x * blockDim.x + threadIdx.x;     // n*16 float4 elements
    if (t < n * (DIM / 4)) {
        int i = t >> 4;                                // node index (16 float4 per row)
        float s = dis[i];
        float s2 = s * s;
        float4 h = H4[t];
        agg4[t] = make_float4(h.x * s2, h.y * s2, h.z * s2, h.w * s2);
    }
}

// edge scatter: agg[dst] += h[src] * dis[src]*dis[dst]
// One wave32 per edge; each lane handles 2 dims (float2) -> the h[src] row is
// one coalesced 256B b64 load per wave; atomics contiguous within the wave.
__global__ void k_scatter(const int* __restrict__ src, const int* __restrict__ dst,
                          const float* __restrict__ H, const float* __restrict__ dis,
                          float* __restrict__ agg, int e) {
    long long t = (long long)blockIdx.x * blockDim.x + threadIdx.x;
    int ed   = (int)(t >> 5);                          // edge = wave
    int lane = (int)(t & 31);
    if (ed < e) {
        int s = src[ed], q = dst[ed];
        float nrm = dis[s] * dis[q];
        v2f h = *(const v2f*)(H + (size_t)s * DIM + lane * 2);
        float* a = agg + (size_t)q * DIM + lane * 2;
        atomicAdd(a + 0, h.x * nrm);
        atomicAdd(a + 1, h.y * nrm);
    }
}

__global__ void k_zeropool(float* __restrict__ gmax, float* __restrict__ gsum) {
    int i = blockIdx.x * blockDim.x + threadIdx.x;
    if (i < NG * DIM) { gmax[i] = 0.0f; gsum[i] = 0.0f; }
}

// x = relu(agg + b); pool max/sum per graph. Post-ReLU values >= 0, so
// atomicMax on the uint bit pattern is order-equivalent to float max.
// float4 per thread for the streaming load/store path.
__global__ void k_relupool(const float4* __restrict__ agg4, const float4* __restrict__ bias4,
                           const int* __restrict__ batch, float4* __restrict__ xout4,
                           float* __restrict__ gmax, float* __restrict__ gsum, int n) {
    int t = blockIdx.x * blockDim.x + threadIdx.x;     // n*16 float4 elements
    if (t < n * (DIM / 4)) {
        int i  = t >> 4;                               // node
        int d4 = t & 15;                               // float4 slot within row
        float4 a = agg4[t];
        float4 b = bias4[d4];
        float4 v = make_float4(fmaxf(a.x + b.x, 0.0f), fmaxf(a.y + b.y, 0.0f),
                               fmaxf(a.z + b.z, 0.0f), fmaxf(a.w + b.w, 0.0f));
        xout4[t] = v;
        int g = batch[i];
        float* gm = gmax + (size_t)g * DIM + d4 * 4;
        float* gs = gsum + (size_t)g * DIM + d4 * 4;
        atomicMax((unsigned int*)(gm + 0), __float_as_uint(v.x));
        atomicMax((unsigned int*)(gm + 1), __float_as_uint(v.y));
        atomicMax((unsigned int*)(gm + 2), __float_as_uint(v.z));
        atomicMax((unsigned int*)(gm + 3), __float_as_uint(v.w));
        atomicAdd(gs + 0, v.x);
        atomicAdd(gs + 1, v.y);
        atomicAdd(gs + 2, v.z);
        atomicAdd(gs + 3, v.w);
    }
}

// graph_emb[g, 0:D] += gmax; graph_emb[g, D:2D] += gsum / count
__global__ void k_poolaccum(const float* __restrict__ gmax, const float* __restrict__ gsum,
                            const float* __restrict__ counts, float* __restrict__ gout) {
    int t = blockIdx.x * blockDim.x + threadIdx.x;
    if (t < NG * DIM) {
        int g = t >> 6;
        int d = t & 63;
        gout[g * 2 * DIM + d]       += gmax[t];
        gout[g * 2 * DIM + DIM + d] += gsum[t] / counts[g];
    }
}

// ---------------------------------------------------------------------------
extern "C" void kernel_launch(void* const* d_in, const int* in_sizes, int n_in,
                              void* d_out, int out_size, void* d_ws, size_t ws_size,
                              hipStream_t stream) {
    const float* x     = (const float*)d_in[0];
    const int*   ei    = (const int*)d_in[1];
    const int*   batch = (const int*)d_in[2];
    const float* Wl[3] = {(const float*)d_in[3], (const float*)d_in[5], (const float*)d_in[7]};
    const float* bl[3] = {(const float*)d_in[4], (const float*)d_in[6], (const float*)d_in[8]};

    const int n = in_sizes[2];          // nodes
    const int e = in_sizes[1] / 2;      // edges
    const int* src = ei;
    const int* dst = ei + e;

    float* out_nodes = (float*)d_out;                     // [n, DIM]
    float* out_graph = out_nodes + (size_t)n * DIM;       // [NG, 2*DIM]

    char* ws = (char*)d_ws;
    float* h      = (float*)ws; ws += (size_t)n * DIM * sizeof(float);
    float* agg    = (float*)ws; ws += (size_t)n * DIM * sizeof(float);
    float* dis    = (float*)ws; ws += (size_t)n * sizeof(float);   // deg -> dis in place
    float* counts = (float*)ws; ws += NG * sizeof(float);
    float* gmax   = (float*)ws; ws += NG * DIM * sizeof(float);
    float* gsum   = (float*)ws; ws += NG * DIM * sizeof(float);

    const int B = 256;
    int initN = (n > 2 * NG * DIM) ? n : 2 * NG * DIM;
    k_init<<<(initN + B - 1) / B, B, 0, stream>>>(dis, counts, out_graph, n);
    k_deg<<<(e + B - 1) / B, B, 0, stream>>>(dst, dis, e);
    k_dis_counts<<<(n + B - 1) / B, B, 0, stream>>>(dis, batch, counts, n);

    const long long nd4 = (long long)n * (DIM / 4);   // float4 elements per N x D pass
    const long long ew  = (long long)e * 32;          // one wave per edge

    const float* xin = x;
    for (int l = 0; l < 3; ++l) {
        k_gemm<<<(n / 16 + 7) / 8, 256, 0, stream>>>(xin, Wl[l], h, n);
        k_self<<<(int)((nd4 + B - 1) / B), B, 0, stream>>>((const float4*)h, dis,
                                                           (float4*)agg, n);
        k_scatter<<<(int)((ew + B - 1) / B), B, 0, stream>>>(src, dst, h, dis, agg, e);
        k_zeropool<<<(NG * DIM + B - 1) / B, B, 0, stream>>>(gmax, gsum);
        k_relupool<<<(int)((nd4 + B - 1) / B), B, 0, stream>>>((const float4*)agg,
                                                               (const float4*)bl[l], batch,
                                                               (float4*)out_nodes,
                                                               gmax, gsum, n);
        k_poolaccum<<<(NG * DIM + B - 1) / B, B, 0, stream>>>(gmax, gsum, counts, out_graph);
        xin = out_nodes;   // next layer reads this layer's activations
    }
}